// MulticlassDiceLossVectorize_56255481643615
// MI455X (gfx1250) — compile-verified
//
#include <hip/hip_runtime.h>

typedef __attribute__((ext_vector_type(2))) float v2f;
typedef __attribute__((ext_vector_type(4))) float v4f;
typedef __attribute__((ext_vector_type(8))) float v8f;

#define NCLS  5
#define EPSF  1e-5f
#define BLOCK 256
#define GRIDX 512

// ---------------------------------------------------------------------------
// Kernel 0: zero the 20-float accumulator workspace (I, Den per (b,c)).
// ---------------------------------------------------------------------------
__global__ void dice_zero_ws(float* __restrict__ ws, int n) {
    const int i = blockIdx.x * blockDim.x + threadIdx.x;
    if (i < n) ws[i] = 0.0f;
}

// ---------------------------------------------------------------------------
// Kernel 1: streaming partial reduction.
//   Per (b,c):  I   = sum x * [t==c]
//               Den = sum (x + [t==c])        (== sum_x + count)
// Each lane accumulates float4 partials (packed f32 VALU), then the wave's
// 32 lanes x v2f partials form one 16x4 f32 A-matrix and a single
// v_wmma_f32_16x16x4_f32 against an all-ones B collapses them to row sums.
// ---------------------------------------------------------------------------
__global__ void __launch_bounds__(BLOCK) dice_partial(
    const float* __restrict__ x, const long long* __restrict__ t,
    float* __restrict__ ws, const long long N) {
    const int b    = blockIdx.y;
    const int lane = (int)(threadIdx.x & 31u);
    const int wavesPerBlock = BLOCK >> 5;
    const long long waveId     = (long long)blockIdx.x * wavesPerBlock + (threadIdx.x >> 5);
    const long long totalWaves = (long long)gridDim.x * wavesPerBlock;
    const long long chunk  = 128;                 // voxels per wave per iteration (4/lane)
    const long long stride = totalWaves * chunk;

    const float*     __restrict__ xb = x + (size_t)b * NCLS * (size_t)N;
    const long long* __restrict__ tb = t + (size_t)b * (size_t)N;

    const v4f vzero = {};
    v4f pInt[NCLS], pDen[NCLS];
#pragma unroll
    for (int c = 0; c < NCLS; ++c) { pInt[c] = vzero; pDen[c] = vzero; }

    // ---- main loop: fully uniform per wave (EXEC stays all-ones) ----------
    const long long nMain = (N / stride) * stride;
    for (long long pos = waveId * chunk; pos < nMain; pos += stride) {
        const long long v = pos + (long long)lane * 4;
        const int4 ta = *reinterpret_cast<const int4*>(tb + v);      // int64 pair: low words .x/.z
        const int4 tc = *reinterpret_cast<const int4*>(tb + v + 2);
        const int t0 = ta.x, t1 = ta.z, t2 = tc.x, t3 = tc.z;
#pragma unroll
        for (int c = 0; c < NCLS; ++c) {
            const v4f xv = *reinterpret_cast<const v4f*>(xb + (size_t)c * (size_t)N + v);
            v4f m = vzero;
            m[0] = (t0 == c) ? 1.0f : 0.0f;
            m[1] = (t1 == c) ? 1.0f : 0.0f;
            m[2] = (t2 == c) ? 1.0f : 0.0f;
            m[3] = (t3 == c) ? 1.0f : 0.0f;
            pDen[c] += xv + m;        // v_pk_add_f32
            pInt[c] += xv * m;        // v_pk_fma_f32
        }
    }

    // ---- scalar tail (empty for the reference shape; kept for generality) -
    const long long gtid     = waveId * 32 + lane;
    const long long nThreads = totalWaves * 32;
    for (long long v = nMain + gtid; v < N; v += nThreads) {
        const int tv = (int)tb[v];
#pragma unroll
        for (int c = 0; c < NCLS; ++c) {
            const float xv = xb[(size_t)c * (size_t)N + v];
            const float m  = (tv == c) ? 1.0f : 0.0f;
            pDen[c][0] += xv + m;
            pInt[c][0] += xv * m;
        }
    }

    __shared__ float sAcc[2 * NCLS];
    if (threadIdx.x < 2 * NCLS) sAcc[threadIdx.x] = 0.0f;
    __syncthreads();

    // ---- wave reduction via the matrix pipe (EXEC all-ones here) ----------
    const v2f onesB = {1.0f, 1.0f};
    const v8f zero8 = {};
    float fI[NCLS], fD[NCLS];
#pragma unroll
    for (int c = 0; c < NCLS; ++c) {
        v2f aI, aD;                                   // fold v4f -> v2f per lane
        aI[0] = pInt[c][0] + pInt[c][2];  aI[1] = pInt[c][1] + pInt[c][3];
        aD[0] = pDen[c][0] + pDen[c][2];  aD[1] = pDen[c][1] + pDen[c][3];
        // D = A(16x4) * ones(4x16): every column of D holds the 16 row sums.
        v8f dI = __builtin_amdgcn_wmma_f32_16x16x4_f32(
            false, aI, false, onesB, (short)0, zero8, false, false);
        v8f dD = __builtin_amdgcn_wmma_f32_16x16x4_f32(
            false, aD, false, onesB, (short)0, zero8, false, false);
        float si = 0.0f, sd = 0.0f;
#pragma unroll
        for (int j = 0; j < 8; ++j) { si += dI[j]; sd += dD[j]; }
        // column 0 lives on lane 0 (rows 0-7) and lane 16 (rows 8-15)
        fI[c] = __shfl(si, 0) + __shfl(si, 16);
        fD[c] = __shfl(sd, 0) + __shfl(sd, 16);
    }
    if (lane == 0) {
#pragma unroll
        for (int c = 0; c < NCLS; ++c) {
            atomicAdd(&sAcc[2 * c + 0], fI[c]);       // ds_add_f32
            atomicAdd(&sAcc[2 * c + 1], fD[c]);
        }
    }
    __syncthreads();
    if (threadIdx.x < 2 * NCLS)
        atomicAdd(&ws[b * (2 * NCLS) + (int)threadIdx.x], sAcc[threadIdx.x]);
}

// ---------------------------------------------------------------------------
// Kernel 2: finalize 10 dice scores -> scalar loss.
// ---------------------------------------------------------------------------
__global__ void dice_finalize(const float* __restrict__ ws, float* __restrict__ out,
                              const int nPairs) {
    if (blockIdx.x == 0 && threadIdx.x == 0) {
        float acc = 0.0f;
        for (int i = 0; i < nPairs; ++i) {
            const float I = ws[2 * i + 0];
            const float D = ws[2 * i + 1];
            acc += (2.0f * I + EPSF) / (D + EPSF);
        }
        out[0] = 1.0f - acc / (float)nPairs;
    }
}

// ---------------------------------------------------------------------------
extern "C" void kernel_launch(void* const* d_in, const int* in_sizes, int n_in,
                              void* d_out, int out_size, void* d_ws, size_t ws_size,
                              hipStream_t stream) {
    const float*     x  = (const float*)d_in[0];       // (B, C, N) fp32
    const long long* tg = (const long long*)d_in[1];   // (B, N) int64
    float* out = (float*)d_out;
    float* ws  = (float*)d_ws;

    const int B = 2;
    const long long N = (long long)in_sizes[1] / B;    // 128^3 = 2,097,152

    dice_zero_ws<<<1, 64, 0, stream>>>(ws, B * NCLS * 2);

    dim3 grid(GRIDX, B, 1);
    dice_partial<<<grid, BLOCK, 0, stream>>>(x, tg, ws, N);

    dice_finalize<<<1, 32, 0, stream>>>(ws, out, B * NCLS);
}